// DifferentialSoftplus_42116449304832
// MI455X (gfx1250) — compile-verified
//
#include <hip/hip_runtime.h>

// out_x = softplus(x)                         [B*n]      f32
// out_dxdq = sigmoid(x)[...,None] * dxdq      [B*n, 128] f32
//
// Roofline: ~1.07 GB streamed, ~0.13 FLOP/byte -> HBM-bound (~46 us floor
// @ 23.3 TB/s). Mapping: 1 wave32 = 1 row of 128 floats (32 lanes x float4
// = 512 B fully-coalesced b128), 4 rows per wave, NT cache policy on the
// streamed tensor (512 MB read > 192 MB L2), global_prefetch_b8 for the
// back half of each chunk. All indexing is 32-bit (tensor < 4 GiB) so the
// compiler can use SADDR+u32-voffset global addressing.

typedef __attribute__((ext_vector_type(4))) float v4f;

#define Q            128u  // inner (q) dimension: 32 lanes * float4
#define ROWS_PER_WAVE  4
#define WAVES_PER_BLK  8   // 256 threads
#define ROWS_PER_BLK  (ROWS_PER_WAVE * WAVES_PER_BLK)

__device__ __forceinline__ void process_row(const float* __restrict__ x,
                                            const float* __restrict__ dxdq,
                                            float* __restrict__ out_x,
                                            float* __restrict__ out_dxdq,
                                            unsigned row, unsigned lane) {
    // Uniform per-row scalar: all lanes load the same x[row] (one cacheline
    // broadcast), share one v_exp_f32 + v_rcp_f32.
    const float xv = x[row];
    const float e  = __expf(-fabsf(xv));              // exp(-|x|), stable
    const float sa = __builtin_amdgcn_rcpf(1.0f + e); // sigmoid(|x|)
    const float sig = (xv >= 0.0f) ? sa : (1.0f - sa);

    if (lane == 0) {
        // softplus(x) = max(x,0) + log1p(exp(-|x|)); the expensive log1p
        // expansion is scoped to lane 0 only (once per row, branched out
        // of the 31-lane hot path).
        out_x[row] = fmaxf(xv, 0.0f) + log1pf(e);
    }

    // 32-bit element offset: max row*Q + 127 = 2^27 - 1 < 2^32.
    const unsigned off = row * Q + lane * 4u;

    // Streamed once, never reused: NT b128 load + store keeps the 1 GB
    // stream from evicting the reused x vector out of L2.
    v4f v = __builtin_nontemporal_load((const v4f*)(dxdq + off));
    v4f o = v * sig;
    __builtin_nontemporal_store(o, (v4f*)(out_dxdq + off));
}

__global__ __launch_bounds__(256)
void diffsoftplus_stream_kernel(const float* __restrict__ x,
                                const float* __restrict__ dxdq,
                                float* __restrict__ out_x,
                                float* __restrict__ out_dxdq,
                                unsigned rows) {
    const unsigned lane  = threadIdx.x & 31u;
    const unsigned waveb = threadIdx.x >> 5;
    const unsigned rowBase =
        (blockIdx.x * WAVES_PER_BLK + waveb) * ROWS_PER_WAVE;

    if (rowBase >= rows) return;

    if (rowBase + ROWS_PER_WAVE <= rows) {
        // Fast path (all full chunks when rows % 32 == 0): no per-row guards.
        // Prefetch the back half of the chunk while the front half streams.
        __builtin_prefetch(dxdq + (rowBase + 2) * Q + lane * 4u, 0, 0);
        __builtin_prefetch(dxdq + (rowBase + 3) * Q + lane * 4u, 0, 0);
#pragma unroll
        for (int r = 0; r < ROWS_PER_WAVE; ++r)
            process_row(x, dxdq, out_x, out_dxdq, rowBase + (unsigned)r, lane);
    } else {
        // Guarded tail (unused for the 2048x512 shape; kept for generality).
        for (unsigned row = rowBase; row < rows; ++row)
            process_row(x, dxdq, out_x, out_dxdq, row, lane);
    }
}

extern "C" void kernel_launch(void* const* d_in, const int* in_sizes, int n_in,
                              void* d_out, int out_size, void* d_ws, size_t ws_size,
                              hipStream_t stream) {
    const float* x    = (const float*)d_in[0];   // [B, n]    flat = rows
    const float* dxdq = (const float*)d_in[1];   // [B, n, q] flat = rows * 128

    const unsigned rows = (unsigned)in_sizes[0]; // 2048*512 = 1,048,576

    float* out_x    = (float*)d_out;             // first `rows` floats
    float* out_dxdq = out_x + (size_t)rows;      // then rows*Q floats

    const unsigned nBlocks = (rows + ROWS_PER_BLK - 1) / ROWS_PER_BLK;
    diffsoftplus_stream_kernel<<<dim3(nBlocks), dim3(256), 0, stream>>>(
        x, dxdq, out_x, out_dxdq, rows);
}